// GATv2WithGlobal_4389456577275
// MI455X (gfx1250) — compile-verified
//
#include <hip/hip_runtime.h>

// ---------------- problem constants (match reference) ----------------
#define NN_NODES 30000
#define EE_EDGES 240000
#define E_TOT    (EE_EDGES + NN_NODES)   // edges + self loops = 270000
#define HC       512
#define NHEAD    4
#define CCH      128
#define F_IN     9
#define G_DIM    50
#define NB_GRAPH 64
#define EPS_BN   1e-5f
#define SLOPE    0.2f

typedef float v8f __attribute__((ext_vector_type(8)));
typedef float v2f __attribute__((ext_vector_type(2)));

// ---------------- helpers ----------------
__device__ __forceinline__ void edge_nodes(const int* __restrict__ ei, int e, int& s, int& d) {
    if (e < EE_EDGES) { s = ei[e]; d = ei[EE_EDGES + e]; }
    else              { s = e - EE_EDGES; d = s; }          // self loop
}

__device__ __forceinline__ void atomicMaxF(float* addr, float val) {
    int* ia = (int*)addr;
    int old = *ia;
    while (__int_as_float(old) < val) {
        int prev = atomicCAS(ia, old, __float_as_int(val));
        if (prev == old) break;
        old = prev;
    }
}

// ---------------- generic fills ----------------
__global__ __launch_bounds__(256) void k_fill(float* __restrict__ p, float v, int n) {
    int i = blockIdx.x * blockDim.x + threadIdx.x;
    if (i < n) p[i] = v;
}

// H[n][j] = bias[j]  (segment_sum output starts at +bias)
__global__ __launch_bounds__(256) void k_init_bias(float* __restrict__ H,
                                                   const float* __restrict__ bias, int n) {
    int i = blockIdx.x * blockDim.x + threadIdx.x;
    if (i < n) H[i] = bias[i & (HC - 1)];
}

// ---------------- layer-1 dual linear: K = 9 ----------------
__global__ __launch_bounds__(256) void k_linear9(const float* __restrict__ x,
                                                 const float* __restrict__ Wl,
                                                 const float* __restrict__ bl,
                                                 const float* __restrict__ Wr,
                                                 const float* __restrict__ br,
                                                 float* __restrict__ XL,
                                                 float* __restrict__ XR) {
    int idx = blockIdx.x * blockDim.x + threadIdx.x;
    if (idx >= NN_NODES * HC) return;
    int n = idx >> 9;
    int j = idx & (HC - 1);
    const float* xr = x + n * F_IN;
    float sl = bl[j], sr = br[j];
#pragma unroll
    for (int k = 0; k < F_IN; ++k) {
        float xv = xr[k];
        sl = fmaf(xv, Wl[k * HC + j], sl);
        sr = fmaf(xv, Wr[k * HC + j], sr);
    }
    XL[idx] = sl;
    XR[idx] = sr;
}

// ---------------- edge attention scores: one wave per (edge, head) ----------------
__global__ __launch_bounds__(256) void k_edge_scores(const float* __restrict__ XL,
                                                     const float* __restrict__ XR,
                                                     const float* __restrict__ att,
                                                     const int* __restrict__ eidx,
                                                     float* __restrict__ scores,
                                                     float* __restrict__ nmax) {
    int wave = (blockIdx.x * blockDim.x + threadIdx.x) >> 5;
    int lane = threadIdx.x & 31;
    int e = wave >> 2;
    int h = wave & 3;
    if (e >= E_TOT) return;
    int s, d;
    edge_nodes(eidx, e, s, d);
    const float* pl = XL + (size_t)s * HC + h * CCH;
    const float* pr = XR + (size_t)d * HC + h * CCH;
    const float* pa = att + h * CCH;
    float acc = 0.f;
#pragma unroll
    for (int i = 0; i < 4; ++i) {
        int c = lane + 32 * i;
        float m = pl[c] + pr[c];
        m = (m > 0.f) ? m : SLOPE * m;
        acc = fmaf(m, pa[c], acc);
    }
#pragma unroll
    for (int off = 16; off > 0; off >>= 1) acc += __shfl_xor(acc, off, 32);
    if (lane == 0) {
        scores[e * NHEAD + h] = acc;
        atomicMaxF(&nmax[d * NHEAD + h], acc);
    }
}

// ---------------- p = exp(e - max[dst]); denom[dst] += p ----------------
__global__ __launch_bounds__(256) void k_edge_exp(float* __restrict__ scores,
                                                  const float* __restrict__ nmax,
                                                  float* __restrict__ denom,
                                                  const int* __restrict__ eidx) {
    int idx = blockIdx.x * blockDim.x + threadIdx.x;
    if (idx >= E_TOT * NHEAD) return;
    int e = idx >> 2;
    int h = idx & 3;
    int s, d;
    edge_nodes(eidx, e, s, d);
    float p = __expf(scores[idx] - nmax[d * NHEAD + h]);
    scores[idx] = p;
    atomicAdd(&denom[d * NHEAD + h], p);
}

// ---------------- out[dst] += alpha * xl[src]: one wave per (edge, head) ----------------
__global__ __launch_bounds__(256) void k_edge_aggregate(const float* __restrict__ scores,
                                                        const float* __restrict__ denom,
                                                        const float* __restrict__ XL,
                                                        const int* __restrict__ eidx,
                                                        float* __restrict__ H) {
    int wave = (blockIdx.x * blockDim.x + threadIdx.x) >> 5;
    int lane = threadIdx.x & 31;
    int e = wave >> 2;
    int h = wave & 3;
    if (e >= E_TOT) return;
    int s, d;
    edge_nodes(eidx, e, s, d);
    float alpha = scores[e * NHEAD + h] / (denom[d * NHEAD + h] + 1e-16f);
    const float* pl = XL + (size_t)s * HC + h * CCH;
    float*       po = H  + (size_t)d * HC + h * CCH;
#pragma unroll
    for (int i = 0; i < 4; ++i) {
        int c = lane + 32 * i;
        atomicAdd(&po[c], alpha * pl[c]);
    }
}

// ---------------- batchnorm: per-channel partial sums (coalesced) ----------------
#define BN_ROWS 60   // 30000 / 60 = 500 blocks
__global__ __launch_bounds__(512) void k_bn_partial(const float* __restrict__ H,
                                                    float* __restrict__ bsum,
                                                    float* __restrict__ bsq) {
    int c = threadIdx.x;
    int r0 = blockIdx.x * BN_ROWS;
    float s = 0.f, q = 0.f;
    for (int r = r0; r < r0 + BN_ROWS; ++r) {
        float v = H[(size_t)r * HC + c];
        s += v;
        q = fmaf(v, v, q);
    }
    atomicAdd(&bsum[c], s);
    atomicAdd(&bsq[c], q);
}

__global__ __launch_bounds__(512) void k_bn_finalize(const float* __restrict__ bsum,
                                                     const float* __restrict__ bsq,
                                                     float* __restrict__ mu,
                                                     float* __restrict__ rstd) {
    int c = threadIdx.x;
    float m = bsum[c] * (1.0f / NN_NODES);
    float var = bsq[c] * (1.0f / NN_NODES) - m * m;   // biased var (train-mode BN)
    mu[c] = m;
    rstd[c] = rsqrtf(var + EPS_BN);
}

__global__ __launch_bounds__(256) void k_bn_apply_relu(float* __restrict__ H,
                                                       const float* __restrict__ mu,
                                                       const float* __restrict__ rstd,
                                                       const float* __restrict__ g,
                                                       const float* __restrict__ b) {
    int idx = blockIdx.x * blockDim.x + threadIdx.x;
    if (idx >= NN_NODES * HC) return;
    int c = idx & (HC - 1);
    float v = (H[idx] - mu[c]) * rstd[c] * g[c] + b[c];
    H[idx] = fmaxf(v, 0.f);
}

// ---------------- layer-2 GEMM via V_WMMA_F32_16X16X4_F32 ----------------
// Y[M,512] = X[M,512] @ W[512,512] + bias.
// Register-blocked: one wave computes a 64x32 output block (4 M-tiles x 2 N-tiles,
// 8 v8f accumulators). Per K-step of 4: 4 b64 A loads + 4 b32 B loads feed 8 WMMAs
// -> 1.0 vmem per wmma (vs 3.0 unblocked). Working set fits the 192MB L2, so the
// kernel is issue-bound and this is the lever that matters.
// A 16x4 f32 fragment (ISA 7.12.2): half = lane>>4, row = lane&15, VGPR r -> K = r + 2*half.
// B 4x16 mirrors it (col = lane&15). C/D: VGPR v, half h -> (M = v + 8h, N = lane&15).
#define MT_TILES (NN_NODES / 16)          // 1875 M-tiles (exact)
#define MB_BLK   ((MT_TILES + 3) / 4)     // 469 M-blocks of 4 tiles (last has 3)
#define NB_BLK   (HC / 32)                // 16 N-blocks of 2 tiles
__global__ __launch_bounds__(256) void k_gemm_wmma_f32(const float* __restrict__ X,
                                                       const float* __restrict__ W,
                                                       const float* __restrict__ bias,
                                                       float* __restrict__ Y) {
    const int K = HC, NNc = HC;
    int wave = (blockIdx.x * blockDim.x + threadIdx.x) >> 5;
    int lane = threadIdx.x & 31;
    int bm = wave / NB_BLK;                 // 0..468
    int bn = wave - bm * NB_BLK;            // 0..15
    if (bm >= MB_BLK) return;
    int half = lane >> 4;
    int r = lane & 15;

    // A row pointers for the 4 M-tiles (clamp invalid tail tiles; stores are guarded)
    const float* Arow[4];
    bool tv[4];
#pragma unroll
    for (int i = 0; i < 4; ++i) {
        int tile = bm * 4 + i;
        tv[i] = (tile < MT_TILES);
        int row = tv[i] ? (tile * 16 + r) : r;
        Arow[i] = X + (size_t)row * K;
    }
    // B column pointers for the 2 N-tiles
    const float* Bcol0 = W + (bn * 32 + r);
    const float* Bcol1 = Bcol0 + 16;

    v8f acc[4][2] = {};
    for (int k = 0; k < K; k += 4) {
        int kb = k + 2 * half;
        v2f a0 = *(const v2f*)(Arow[0] + kb);
        v2f a1 = *(const v2f*)(Arow[1] + kb);
        v2f a2 = *(const v2f*)(Arow[2] + kb);
        v2f a3 = *(const v2f*)(Arow[3] + kb);
        v2f b0, b1;
        b0.x = Bcol0[(size_t)kb * NNc];
        b0.y = Bcol0[(size_t)(kb + 1) * NNc];
        b1.x = Bcol1[(size_t)kb * NNc];
        b1.y = Bcol1[(size_t)(kb + 1) * NNc];
        acc[0][0] = __builtin_amdgcn_wmma_f32_16x16x4_f32(false, a0, false, b0, (short)0, acc[0][0], false, false);
        acc[0][1] = __builtin_amdgcn_wmma_f32_16x16x4_f32(false, a0, false, b1, (short)0, acc[0][1], false, false);
        acc[1][0] = __builtin_amdgcn_wmma_f32_16x16x4_f32(false, a1, false, b0, (short)0, acc[1][0], false, false);
        acc[1][1] = __builtin_amdgcn_wmma_f32_16x16x4_f32(false, a1, false, b1, (short)0, acc[1][1], false, false);
        acc[2][0] = __builtin_amdgcn_wmma_f32_16x16x4_f32(false, a2, false, b0, (short)0, acc[2][0], false, false);
        acc[2][1] = __builtin_amdgcn_wmma_f32_16x16x4_f32(false, a2, false, b1, (short)0, acc[2][1], false, false);
        acc[3][0] = __builtin_amdgcn_wmma_f32_16x16x4_f32(false, a3, false, b0, (short)0, acc[3][0], false, false);
        acc[3][1] = __builtin_amdgcn_wmma_f32_16x16x4_f32(false, a3, false, b1, (short)0, acc[3][1], false, false);
    }

    float bj0 = bias[bn * 32 + r];
    float bj1 = bias[bn * 32 + 16 + r];
#pragma unroll
    for (int i = 0; i < 4; ++i) {
        if (!tv[i]) continue;
        float* Yt = Y + (size_t)((bm * 4 + i) * 16) * NNc + bn * 32 + r;
#pragma unroll
        for (int v = 0; v < 8; ++v) {
            size_t roff = (size_t)(v + 8 * half) * NNc;
            Yt[roff]      = acc[i][0][v] + bj0;
            Yt[roff + 16] = acc[i][1][v] + bj1;
        }
    }
}

// ---------------- global mean pool (accumulate) ----------------
__global__ __launch_bounds__(256) void k_pool_acc(const float* __restrict__ H,
                                                  const int* __restrict__ batch,
                                                  float* __restrict__ pooled,
                                                  float* __restrict__ counts) {
    int idx = blockIdx.x * blockDim.x + threadIdx.x;
    if (idx >= NN_NODES * HC) return;
    int n = idx >> 9;
    int c = idx & (HC - 1);
    int b = batch[n];
    atomicAdd(&pooled[b * HC + c], H[idx]);
    if (c == 0) atomicAdd(&counts[b], 1.0f);
}

// ---------------- MLP head: one block per graph ----------------
__global__ __launch_bounds__(128) void k_mlp_head(const float* __restrict__ pooled,
                                                  const float* __restrict__ counts,
                                                  const float* __restrict__ gfeat,
                                                  const float* __restrict__ fc1w,
                                                  const float* __restrict__ fc1b,
                                                  const float* __restrict__ fc2w,
                                                  const float* __restrict__ fc2b,
                                                  float* __restrict__ out) {
    __shared__ float sp[HC];
    __shared__ float sg[G_DIM];
    __shared__ float red[128];
    int b = blockIdx.x;
    int j = threadIdx.x;
    float inv = 1.0f / fmaxf(counts[b], 1.0f);
    for (int k = j; k < HC; k += 128) sp[k] = pooled[b * HC + k] * inv;
    if (j < G_DIM) sg[j] = gfeat[b * G_DIM + j];
    __syncthreads();
    float z = fc1b[j];
    for (int k = 0; k < HC; ++k)    z = fmaf(sp[k], fc1w[k * CCH + j], z);
    for (int k = 0; k < G_DIM; ++k) z = fmaf(sg[k], fc1w[(HC + k) * CCH + j], z);
    z = fmaxf(z, 0.f);
    red[j] = z * fc2w[j];
    __syncthreads();
    for (int s = 64; s > 0; s >>= 1) {
        if (j < s) red[j] += red[j + s];
        __syncthreads();
    }
    if (j == 0) out[b] = red[0] + fc2b[0];
}

// ---------------- host orchestration ----------------
static void run_gat_edges(const float* XL, const float* XR, const float* att,
                          const int* eidx, float* scores, float* nmax, float* denom,
                          float* H, const float* bias, hipStream_t stream) {
    const int NHC = NN_NODES * HC;
    k_init_bias<<<(NHC + 255) / 256, 256, 0, stream>>>(H, bias, NHC);
    k_fill<<<(NN_NODES * NHEAD + 255) / 256, 256, 0, stream>>>(nmax, -__builtin_huge_valf(), NN_NODES * NHEAD);
    k_fill<<<(NN_NODES * NHEAD + 255) / 256, 256, 0, stream>>>(denom, 0.f, NN_NODES * NHEAD);
    int waves = E_TOT * NHEAD;                      // one wave per (edge, head)
    int blocks = (waves + 7) / 8;                   // 256 threads = 8 waves
    k_edge_scores<<<blocks, 256, 0, stream>>>(XL, XR, att, eidx, scores, nmax);
    k_edge_exp<<<(E_TOT * NHEAD + 255) / 256, 256, 0, stream>>>(scores, nmax, denom, eidx);
    k_edge_aggregate<<<blocks, 256, 0, stream>>>(scores, denom, XL, eidx, H);
}

static void run_bn_relu(float* H, const float* g, const float* b,
                        float* bsum, float* bsq, float* mu, float* rstd,
                        hipStream_t stream) {
    const int NHC = NN_NODES * HC;
    k_fill<<<(HC * 2 + 255) / 256, 256, 0, stream>>>(bsum, 0.f, HC * 2); // bsum+bsq contiguous
    k_bn_partial<<<NN_NODES / BN_ROWS, 512, 0, stream>>>(H, bsum, bsq);
    k_bn_finalize<<<1, 512, 0, stream>>>(bsum, bsq, mu, rstd);
    k_bn_apply_relu<<<(NHC + 255) / 256, 256, 0, stream>>>(H, mu, rstd, g, b);
}

extern "C" void kernel_launch(void* const* d_in, const int* in_sizes, int n_in,
                              void* d_out, int out_size, void* d_ws, size_t ws_size,
                              hipStream_t stream) {
    const float* x     = (const float*)d_in[0];
    const float* gfeat = (const float*)d_in[1];
    const float* W1l = (const float*)d_in[2];
    const float* b1l = (const float*)d_in[3];
    const float* W1r = (const float*)d_in[4];
    const float* b1r = (const float*)d_in[5];
    const float* att1  = (const float*)d_in[6];
    const float* bias1 = (const float*)d_in[7];
    const float* W2l = (const float*)d_in[8];
    const float* b2l = (const float*)d_in[9];
    const float* W2r = (const float*)d_in[10];
    const float* b2r = (const float*)d_in[11];
    const float* att2  = (const float*)d_in[12];
    const float* bias2 = (const float*)d_in[13];
    const float* bn1g = (const float*)d_in[14];
    const float* bn1b = (const float*)d_in[15];
    const float* bn2g = (const float*)d_in[16];
    const float* bn2b = (const float*)d_in[17];
    const float* fc1w = (const float*)d_in[18];
    const float* fc1b = (const float*)d_in[19];
    const float* fc2w = (const float*)d_in[20];
    const float* fc2b = (const float*)d_in[21];
    const int*   eidx  = (const int*)d_in[22];
    const int*   batch = (const int*)d_in[23];
    float* out = (float*)d_out;

    // ---- workspace layout (floats) ----
    float* ws = (float*)d_ws;
    const size_t NB = (size_t)NN_NODES * HC;        // 15.36M floats
    float* XL     = ws;
    float* XR     = XL + NB;
    float* H1     = XR + NB;
    float* H2     = H1 + NB;
    float* scores = H2 + NB;                        // E_TOT*4
    float* nmax   = scores + (size_t)E_TOT * NHEAD; // 120k
    float* denom  = nmax + (size_t)NN_NODES * NHEAD;
    float* bsum   = denom + (size_t)NN_NODES * NHEAD; // 512
    float* bsq    = bsum + HC;                      // 512 (contiguous w/ bsum)
    float* mu     = bsq + HC;
    float* rstd   = mu + HC;
    float* pooled = rstd + HC;                      // 64*512
    float* counts = pooled + (size_t)NB_GRAPH * HC; // 64

    const int NHC = NN_NODES * HC;

    // ---- layer 1 ----
    k_linear9<<<(NHC + 255) / 256, 256, 0, stream>>>(x, W1l, b1l, W1r, b1r, XL, XR);
    run_gat_edges(XL, XR, att1, eidx, scores, nmax, denom, H1, bias1, stream);
    run_bn_relu(H1, bn1g, bn1b, bsum, bsq, mu, rstd, stream);

    // ---- layer 2: WMMA fp32 GEMMs for both transforms ----
    {
        int waves = MB_BLK * NB_BLK;                // 469 * 16 = 7504 waves
        int blocks = (waves + 7) / 8;
        k_gemm_wmma_f32<<<blocks, 256, 0, stream>>>(H1, W2l, b2l, XL);
        k_gemm_wmma_f32<<<blocks, 256, 0, stream>>>(H1, W2r, b2r, XR);
    }
    run_gat_edges(XL, XR, att2, eidx, scores, nmax, denom, H2, bias2, stream);
    run_bn_relu(H2, bn2g, bn2b, bsum, bsq, mu, rstd, stream);

    // ---- pool + head ----
    k_fill<<<(NB_GRAPH * HC + NB_GRAPH + 255) / 256, 256, 0, stream>>>(pooled, 0.f, NB_GRAPH * HC + NB_GRAPH); // pooled + counts contiguous
    k_pool_acc<<<(NHC + 255) / 256, 256, 0, stream>>>(H2, batch, pooled, counts);
    k_mlp_head<<<NB_GRAPH, 128, 0, stream>>>(pooled, counts, gfeat, fc1w, fc1b, fc2w, fc2b, out);
}